// SileroVAD16k_5239860101461
// MI455X (gfx1250) — compile-verified
//
#include <hip/hip_runtime.h>
#include <hip/hip_bf16.h>

typedef float v2f __attribute__((ext_vector_type(2)));
typedef float v8f __attribute__((ext_vector_type(8)));

#define B_TOTAL 16384
#define WAVES   4

// ---------------- ws layout (floats): repacked, zero-padded B matrices ----------------
#define OFF_B0   0         // STFT  B0[k=256][n=272]   (n<258 valid)
#define OFF_B1   69632     // conv1 B1[dt=3][c=132][n=128] (c<129 valid)
#define OFF_B2   120320    // conv2 B2[dt=3][c=132][n=64]  (c<128 valid)
#define OFF_B3   145664    // conv3 B3[dt=3][c=68][n=64]   (c<64 valid)
#define OFF_B4   158720    // conv4 B4[dt=3][c=68][n=128]  (c<64 valid)
#define OFF_B5   184832    // lstm  B5[k=256][n=512] = [w_ih.T ; w_hh.T]
#define OFF_BI5  315904    // combined bias b_ih + b_hh [512]
#define WS_TOTAL_F 316416

// ---------------- per-wave LDS arena (floats); strides are 4*odd -> conflict-free ------
#define R1_F 6336          // 25,344 B : hbuf -> c2buf -> lstmin -> hnbuf
#define R2_F 8256          // 33,024 B : sbuf -> c1buf -> c3buf -> gatebuf
#define ARENA_F (R1_F + R2_F)
#define S_SB 276   // stft raw   [16][276]
#define S_HB 396   // magnitude  [16][3][132]
#define S_C1 396   // conv1 out  [16][3][132]
#define S_C2 140   // conv2 out  [16][2][68]+pad
#define S_C3 68    // conv3 out  [16][68]
#define S_LI 260   // lstm input [16][260] (feat 0..127, h_state 128..255)
#define S_GB 516   // gates      [16][516]
#define S_HN 132   // hn staging [16][132]

__device__ __forceinline__ v8f wmma4(v2f a, v2f b, v8f c) {
  // V_WMMA_F32_16X16X4_F32 : D = A(16x4,f32) * B(4x16,f32) + C(16x16,f32)
  return __builtin_amdgcn_wmma_f32_16x16x4_f32(false, a, false, b, (short)0, c, false, false);
}

__device__ __forceinline__ float fsig(float x) { return 1.0f / (1.0f + __expf(-x)); }

template <int NT>
__device__ __forceinline__ void zero_acc(v8f acc[NT]) {
#pragma unroll
  for (int j = 0; j < NT; ++j)
#pragma unroll
    for (int v = 0; v < 8; ++v) acc[j][v] = 0.0f;
}

// ---- software-pipelined GEMM inner loop: A row in LDS, B block in global [K][N] -------
// A-frag layout: lanes 0-15 hold K=k0,k0+1 ; lanes 16-31 hold K=k0+2,k0+3.
// Loads for step k0+4 are issued before the WMMAs of step k0 (hides L2 latency).
template <int NT>
__device__ __forceinline__ void gemm_rows(const float* __restrict__ Arow,
                                          const float* __restrict__ Bblk,
                                          int K, int N, int nbase, int kh, int nB,
                                          v8f acc[NT]) {
  __builtin_prefetch(Bblk + nbase + nB, 0, 1);   // global_prefetch_b8
  v2f a_cur = *(const v2f*)(Arow + kh);
  const float* bp = Bblk + kh * N + nbase + nB;
  v2f b_cur[NT];
#pragma unroll
  for (int j = 0; j < NT; ++j) { b_cur[j].x = bp[j * 16]; b_cur[j].y = bp[j * 16 + N]; }
#pragma unroll 2
  for (int k0 = 0; k0 < K - 4; k0 += 4) {
    const float* bq = bp + 4 * N;
    v2f a_nxt = *(const v2f*)(Arow + k0 + 4 + kh);
    v2f b_nxt[NT];
#pragma unroll
    for (int j = 0; j < NT; ++j) { b_nxt[j].x = bq[j * 16]; b_nxt[j].y = bq[j * 16 + N]; }
#pragma unroll
    for (int j = 0; j < NT; ++j) acc[j] = wmma4(a_cur, b_cur[j], acc[j]);
    a_cur = a_nxt;
#pragma unroll
    for (int j = 0; j < NT; ++j) b_cur[j] = b_nxt[j];
    bp = bq;
  }
#pragma unroll
  for (int j = 0; j < NT; ++j) acc[j] = wmma4(a_cur, b_cur[j], acc[j]);
}

// ---- STFT GEMM: A fragments come from global x with reflect-pad index math -----------
template <int NT>
__device__ __forceinline__ void stft_gemm(const float* __restrict__ xrow, int w,
                                          const float* __restrict__ B0, int nbase,
                                          int kh, int nB, v8f acc[NT]) {
  const int ib = w * 128 + kh;
  v2f a_cur;
  { int i0 = ib, i1 = ib + 1;
    a_cur.x = xrow[i0 < 512 ? i0 : 1022 - i0];
    a_cur.y = xrow[i1 < 512 ? i1 : 1022 - i1]; }
  const float* bp = B0 + kh * 272 + nbase + nB;
  v2f b_cur[NT];
#pragma unroll
  for (int j = 0; j < NT; ++j) { b_cur[j].x = bp[j * 16]; b_cur[j].y = bp[j * 16 + 272]; }
#pragma unroll 2
  for (int k0 = 0; k0 < 252; k0 += 4) {
    const float* bq = bp + 4 * 272;
    v2f a_nxt;
    { int i0 = ib + k0 + 4, i1 = i0 + 1;
      a_nxt.x = xrow[i0 < 512 ? i0 : 1022 - i0];
      a_nxt.y = xrow[i1 < 512 ? i1 : 1022 - i1]; }
    v2f b_nxt[NT];
#pragma unroll
    for (int j = 0; j < NT; ++j) { b_nxt[j].x = bq[j * 16]; b_nxt[j].y = bq[j * 16 + 272]; }
#pragma unroll
    for (int j = 0; j < NT; ++j) acc[j] = wmma4(a_cur, b_cur[j], acc[j]);
    a_cur = a_nxt;
#pragma unroll
    for (int j = 0; j < NT; ++j) b_cur[j] = b_nxt[j];
    bp = bq;
  }
#pragma unroll
  for (int j = 0; j < NT; ++j) acc[j] = wmma4(a_cur, b_cur[j], acc[j]);
}

// C/D layout: VGPR v holds M = v + 8*(lane>=16), N = (lane&15)+16*tile
template <int NT>
__device__ __forceinline__ void store_tile(float* __restrict__ dst, int strideM,
                                           const float* __restrict__ bias, bool do_relu,
                                           const v8f acc[NT], int nbase, int lane) {
  const int hi = (lane >> 4) << 3;
  const int nB = lane & 15;
#pragma unroll
  for (int j = 0; j < NT; ++j) {
    const int n = nbase + j * 16 + nB;
    const float bv = bias ? bias[n] : 0.0f;
#pragma unroll
    for (int v = 0; v < 8; ++v) {
      float val = acc[j][v] + bv;
      if (do_relu) val = fmaxf(val, 0.0f);
      dst[(v + hi) * strideM + n] = val;
    }
  }
}

// ---------------------------------------------------------------------------------------
__global__ void vad_repack(const float* __restrict__ stft_w, const float* __restrict__ w1,
                           const float* __restrict__ w2, const float* __restrict__ w3,
                           const float* __restrict__ w4, const float* __restrict__ w_ih,
                           const float* __restrict__ w_hh, const float* __restrict__ b_ih,
                           const float* __restrict__ b_hh, float* __restrict__ ws) {
  for (int idx = blockIdx.x * blockDim.x + threadIdx.x; idx < WS_TOTAL_F;
       idx += gridDim.x * blockDim.x) {
    float v = 0.0f;
    if (idx < OFF_B1) {                       // B0: [k][n] = stft_w[n,0,k]
      int k = idx / 272, n = idx % 272;
      if (n < 258) v = stft_w[n * 256 + k];
    } else if (idx < OFF_B2) {                // B1: [dt][c][n] = w1[n,c,dt]
      int r = idx - OFF_B1; int dt = r / (132 * 128); int c = (r / 128) % 132; int n = r & 127;
      if (c < 129) v = w1[(n * 129 + c) * 3 + dt];
    } else if (idx < OFF_B3) {
      int r = idx - OFF_B2; int dt = r / (132 * 64); int c = (r / 64) % 132; int n = r & 63;
      if (c < 128) v = w2[(n * 128 + c) * 3 + dt];
    } else if (idx < OFF_B4) {
      int r = idx - OFF_B3; int dt = r / (68 * 64); int c = (r / 64) % 68; int n = r & 63;
      if (c < 64) v = w3[(n * 64 + c) * 3 + dt];
    } else if (idx < OFF_B5) {
      int r = idx - OFF_B4; int dt = r / (68 * 128); int c = (r / 128) % 68; int n = r & 127;
      if (c < 64) v = w4[(n * 64 + c) * 3 + dt];
    } else if (idx < OFF_BI5) {               // B5: [k][n] = k<128 ? w_ih[n,k] : w_hh[n,k-128]
      int r = idx - OFF_B5; int k = r / 512; int n = r & 511;
      v = (k < 128) ? w_ih[n * 128 + k] : w_hh[n * 128 + (k - 128)];
    } else {
      int n = idx - OFF_BI5;
      v = b_ih[n] + b_hh[n];
    }
    ws[idx] = v;
  }
}

// ---------------------------------------------------------------------------------------
__global__ __launch_bounds__(32 * WAVES, 1)
void vad_fused(const float* __restrict__ x, const float* __restrict__ state,
               const float* __restrict__ b1, const float* __restrict__ b2,
               const float* __restrict__ b3, const float* __restrict__ b4,
               const float* __restrict__ fw, const float* __restrict__ fb,
               const float* __restrict__ ws, float* __restrict__ out) {
  extern __shared__ float smem[];
  const int wid  = threadIdx.x >> 5;
  const int lane = threadIdx.x & 31;
  float* R1 = smem + wid * ARENA_F;
  float* R2 = R1 + R1_F;
  const int r0 = (blockIdx.x * WAVES + wid) * 16;   // this wave's 16 batch rows

  const int mA = lane & 15;          // A-matrix row (batch)
  const int kh = (lane >> 4) << 1;   // K sub-offset 0 / 2
  const int nB = lane & 15;          // B/D column within tile
  const int hi = lane >> 4;

  const float* B0 = ws + OFF_B0;  const float* B1m = ws + OFF_B1;
  const float* B2m = ws + OFF_B2; const float* B3m = ws + OFF_B3;
  const float* B4m = ws + OFF_B4; const float* B5m = ws + OFF_B5;
  const float* bi5 = ws + OFF_BI5;

  // ============ Phase 1: STFT + magnitude (3 windows) ============
  float* sbuf = R2;
  float* hbuf = R1;
  const float* xrow = x + (size_t)(r0 + mA) * 512;
  for (int w = 0; w < 3; ++w) {
    for (int nc0 = 0; nc0 < 16; nc0 += 4) {             // 16 full tiles
      v8f acc[4]; zero_acc<4>(acc);
      stft_gemm<4>(xrow, w, B0, nc0 * 16, kh, nB, acc);
      store_tile<4>(sbuf, S_SB, nullptr, false, acc, nc0 * 16, lane);
    }
    {                                                    // tail tile (n 256..271)
      v8f acc1[1]; zero_acc<1>(acc1);
      stft_gemm<1>(xrow, w, B0, 256, kh, nB, acc1);
      store_tile<1>(sbuf, S_SB, nullptr, false, acc1, 256, lane);
    }
    __syncthreads();
    for (int idx = lane; idx < 16 * 132; idx += 32) {    // |real+i*imag| -> hbuf[:, w, :]
      int m = idx / 132, c = idx % 132;
      float val = 0.0f;
      if (c < 129) {
        float re = sbuf[m * S_SB + c], im = sbuf[m * S_SB + 129 + c];
        val = sqrtf(re * re + im * im);
      }
      hbuf[m * S_HB + w * 132 + c] = val;
    }
    __syncthreads();
  }

  // ============ Phase 2: conv1 (129->128, k3, s1, p1, relu) ============
  float* c1 = R2;
  for (int t_out = 0; t_out < 3; ++t_out)
    for (int nc0 = 0; nc0 < 8; nc0 += 4) {
      v8f acc[4]; zero_acc<4>(acc);
      for (int dt = 0; dt < 3; ++dt) {
        int tin = t_out + dt - 1;
        if (tin < 0 || tin > 2) continue;
        gemm_rows<4>(hbuf + mA * S_HB + tin * 132, B1m + dt * 132 * 128, 132, 128,
                     nc0 * 16, kh, nB, acc);
      }
      store_tile<4>(c1 + t_out * 132, S_C1, b1, true, acc, nc0 * 16, lane);
    }
  for (int idx = lane; idx < 16 * 3 * 4; idx += 32) {    // zero channel pad 128..131
    int m = idx / 12, r = idx % 12;
    c1[m * S_C1 + (r >> 2) * 132 + 128 + (r & 3)] = 0.0f;
  }
  __syncthreads();

  // ============ Phase 3: conv2 (128->64, k3, s2, p1, relu) ============
  float* c2 = R1;
  for (int t_out = 0; t_out < 2; ++t_out) {
    v8f acc[4]; zero_acc<4>(acc);
    for (int dt = 0; dt < 3; ++dt) {
      int tin = 2 * t_out + dt - 1;
      if (tin < 0 || tin > 2) continue;
      gemm_rows<4>(c1 + mA * S_C1 + tin * 132, B2m + dt * 132 * 64, 132, 64, 0, kh, nB, acc);
    }
    store_tile<4>(c2 + t_out * 68, S_C2, b2, true, acc, 0, lane);
  }
  for (int idx = lane; idx < 16 * 2 * 4; idx += 32) {
    int m = idx / 8, r = idx % 8;
    c2[m * S_C2 + (r >> 2) * 68 + 64 + (r & 3)] = 0.0f;
  }
  __syncthreads();

  // ============ Phase 4: conv3 (64->64, k3, s2, p1, relu) ============
  float* c3 = R2;
  {
    v8f acc[4]; zero_acc<4>(acc);
    for (int dt = 1; dt < 3; ++dt)                       // tin = dt-1 in {0,1}
      gemm_rows<4>(c2 + mA * S_C2 + (dt - 1) * 68, B3m + dt * 68 * 64, 68, 64, 0, kh, nB, acc);
    store_tile<4>(c3, S_C3, b3, true, acc, 0, lane);
  }
  for (int idx = lane; idx < 16 * 4; idx += 32) c3[(idx >> 2) * S_C3 + 64 + (idx & 3)] = 0.0f;
  __syncthreads();

  // ============ Phase 5: conv4 (64->128, only dt=1 valid) + gather LSTM input ============
  float* li = R1;
  for (int nc0 = 0; nc0 < 8; nc0 += 4) {
    v8f acc[4]; zero_acc<4>(acc);
    gemm_rows<4>(c3 + mA * S_C3, B4m + 1 * 68 * 128, 68, 128, nc0 * 16, kh, nB, acc);
    store_tile<4>(li, S_LI, b4, true, acc, nc0 * 16, lane);  // feat -> li[:,0:128]
  }
  for (int idx = lane; idx < 16 * 128; idx += 32) {      // h_state -> li[:,128:256]
    int m = idx >> 7, c = idx & 127;
    li[m * S_LI + 128 + c] = state[(size_t)(r0 + m) * 128 + c];
  }
  __syncthreads();

  // ============ Phase 6: LSTM gates GEMM  [16x256]x[256x512] ============
  float* gb = R2;
  for (int nc0 = 0; nc0 < 32; nc0 += 4) {
    v8f acc[4]; zero_acc<4>(acc);
    gemm_rows<4>(li + mA * S_LI, B5m, 256, 512, nc0 * 16, kh, nB, acc);
    store_tile<4>(gb, S_GB, bi5, false, acc, nc0 * 16, lane);
  }
  __syncthreads();

  // ============ Phase 7: LSTM pointwise + state out ============
  float* hnb = R1;
  const float* c_old = state + (size_t)B_TOTAL * 128;
  float* out_hn = out + B_TOTAL;
  float* out_c  = out + B_TOTAL + (size_t)B_TOTAL * 128;
  for (int idx = lane; idx < 16 * 128; idx += 32) {
    int m = idx >> 7, c = idx & 127;
    float ig = gb[m * S_GB + c];
    float fg = gb[m * S_GB + 128 + c];
    float gg = gb[m * S_GB + 256 + c];
    float og = gb[m * S_GB + 384 + c];
    float co = c_old[(size_t)(r0 + m) * 128 + c];
    float cn = fsig(fg) * co + fsig(ig) * tanhf(gg);
    float hn = fsig(og) * tanhf(cn);
    out_hn[(size_t)(r0 + m) * 128 + c] = hn;
    out_c [(size_t)(r0 + m) * 128 + c] = cn;
    hnb[m * S_HN + c] = hn;
  }
  __syncthreads();

  // ============ Phase 8: sigmoid(relu(hn)·fw + fb) head, wave32 pair reduction ============
  {
    int m = lane >> 1, p = lane & 1;
    float s = 0.0f;
    for (int j = 0; j < 64; ++j) {
      int c = p * 64 + j;
      s += fmaxf(hnb[m * S_HN + c], 0.0f) * fw[c];
    }
    s += __shfl_xor(s, 1, 32);
    if (p == 0) out[r0 + m] = fsig(s + fb[0]);
  }
}

// ---------------------------------------------------------------------------------------
extern "C" void kernel_launch(void* const* d_in, const int* in_sizes, int n_in,
                              void* d_out, int out_size, void* d_ws, size_t ws_size,
                              hipStream_t stream) {
  const float* x      = (const float*)d_in[0];
  const float* state  = (const float*)d_in[1];
  const float* stft_w = (const float*)d_in[2];
  const float* w1 = (const float*)d_in[3];  const float* b1 = (const float*)d_in[4];
  const float* w2 = (const float*)d_in[5];  const float* b2 = (const float*)d_in[6];
  const float* w3 = (const float*)d_in[7];  const float* b3 = (const float*)d_in[8];
  const float* w4 = (const float*)d_in[9];  const float* b4 = (const float*)d_in[10];
  const float* w_ih = (const float*)d_in[11]; const float* w_hh = (const float*)d_in[12];
  const float* b_ih = (const float*)d_in[13]; const float* b_hh = (const float*)d_in[14];
  const float* fw = (const float*)d_in[15];   const float* fb = (const float*)d_in[16];
  float* out = (float*)d_out;
  float* ws  = (float*)d_ws;

  vad_repack<<<512, 256, 0, stream>>>(stft_w, w1, w2, w3, w4, w_ih, w_hh, b_ih, b_hh, ws);

  const int blocks = B_TOTAL / (WAVES * 16);                       // 256 workgroups
  const size_t smem = (size_t)WAVES * ARENA_F * sizeof(float);     // 233,472 B of LDS
  vad_fused<<<blocks, WAVES * 32, smem, stream>>>(x, state, b1, b2, b3, b4, fw, fb, ws, out);
}